// VN_PointNet_PSEG_mac_63007170232725
// MI455X (gfx1250) — compile-verified
//
#include <hip/hip_runtime.h>
#include <hip/hip_bf16.h>
#include <math.h>

// ---------------------------------------------------------------------------
// VN-PointNet part-seg forward for gfx1250 (MI455X).
// All channel-mixing matmuls run on the CDNA5 matrix pipe via
// V_WMMA_F32_16X16X4_F32 (fp32 in / fp32 acc, keeps reference precision),
// with LDS-staged 64x128 block tiles, K-paired v2f LDS layout so every
// fragment is a single ds_load_b64, b128 global staging, and branchless
// K-tail handling (EXEC all-ones at every WMMA).
// ---------------------------------------------------------------------------

#define EPSV 1e-6f
#define BNE  1e-5f

typedef __attribute__((ext_vector_type(2))) float v2f;
typedef __attribute__((ext_vector_type(4))) float v4f;
typedef __attribute__((ext_vector_type(8))) float v8f;

static constexpr int  Bn = 4;
static constexpr int  Nn = 2048;
static constexpr int  Kn = 20;
static constexpr int  Mv = Bn * 3 * Nn;             // 24576  (B*3*N)
static constexpr long MkL = (long)Bn * 3 * Nn * Kn; // 491520 (B*3*N*K)
static constexpr int  Mc = Bn * Nn;                 // 8192   (B*N)

// ---------------------------------------------------------------------------
// GEMM: C[Co x M] = A[Co x Kd] * X[Kd x M]   (all row-major, fp32)
//
// Block = 256 threads = 8 waves arranged 4(row) x 2(col).
// Block tile = 64 rows x 128 cols; each wave owns FOUR 16x16 C tiles
// (16 rows x 64 cols) -> 32 WMMAs per wave per 32-K chunk.
//
// LDS is stored K-PAIRED as v2f: element [kp][r] = (val(k=2kp), val(k=2kp+1)).
// A lane's fragment (.x = K, .y = K+1 of its K-half) is then exactly one
// aligned ds_load_b64 into a contiguous VGPR pair -> no register shuffles.
//   A fragment j: As2[2j + kph][wrow*16 + mrow]   (kph = lane>=16)
//   B fragment j: Xs2[2j + kph][wcol*64 + s*16 + mrow]
//
// Bank windows (64 banks, 8B accesses): Xs2 pitch 144 -> pair-row stride
// 288 words = +32 banks: half-wave windows disjoint; staging stores hit 32
// distinct even banks.  As2 pitch 81 -> stores all-distinct banks
// (2*(17q mod 32)+comp), reads overlap on 2/32 banks only.
//
// Correctness-by-clamping: OOB rows/cols load from clamped valid addresses
// and only pollute C entries never stored.  Only the K-tail needs true
// zeros, done with value selects (v_cndmask), never EXEC-divergent branches,
// so EXEC is all-ones at every WMMA.
// ---------------------------------------------------------------------------
__global__ __launch_bounds__(256) void gemm_wmma_f32(
    const float* __restrict__ A, const float* __restrict__ X,
    float* __restrict__ C, int Co, int Kd, int M)
{
    __shared__ v2f As2[16][81];     // [kpair][row]
    __shared__ v2f Xs2[16][144];    // [kpair][col]

    const int tid   = threadIdx.x;
    const int lane  = tid & 31;
    const int wave  = tid >> 5;
    const int wrow  = wave >> 1;          // 0..3
    const int wcol  = wave & 1;           // 0..1
    const int rblk  = blockIdx.y * 64;
    const int cblk  = blockIdx.x * 128;
    const int mrow  = lane & 15;
    const int kph   = lane >> 4;          // 0 or 1 (K-half / pair select)

    v8f acc0 = {}, acc1 = {}, acc2 = {}, acc3 = {};

    for (int k0 = 0; k0 < Kd; k0 += 32) {
        __syncthreads();
        // stage A chunk: 64 rows x 32 k, coalesced along k
        #pragma unroll
        for (int i = 0; i < 8; ++i) {
            int e   = tid + i * 256;
            int row = e >> 5, kk = e & 31;
            int ra  = rblk + row;  ra = (ra < Co) ? ra : (Co - 1);
            int kg  = k0 + kk;
            int kc  = (kg < Kd) ? kg : (Kd - 1);
            float v = A[(size_t)ra * Kd + kc];
            ((float*)&As2[kk >> 1][row])[kk & 1] = (kg < Kd) ? v : 0.f;
        }
        // stage X chunk: 32 k x 128 cols
        if ((M & 3) == 0) {
            // vector path: 4x b128 loads per thread (rows 16B-aligned)
            #pragma unroll
            for (int i = 0; i < 4; ++i) {
                int e    = tid + i * 256;
                int krow = e >> 5;
                int col4 = (e & 31) * 4;
                int kg   = k0 + krow;
                int kc   = (kg < Kd) ? kg : (Kd - 1);
                int cg   = cblk + col4;
                cg = (cg <= M - 4) ? cg : (M - 4);
                v4f vv = *(const v4f*)(X + (size_t)kc * M + cg);
                bool ok = (kg < Kd);
                float* dst = ((float*)&Xs2[krow >> 1][col4]) + (krow & 1);
                dst[0] = ok ? vv[0] : 0.f;
                dst[2] = ok ? vv[1] : 0.f;
                dst[4] = ok ? vv[2] : 0.f;
                dst[6] = ok ? vv[3] : 0.f;
            }
        } else {
            // scalar path (M not a multiple of 4)
            #pragma unroll
            for (int i = 0; i < 16; ++i) {
                int e    = tid + i * 256;
                int krow = e >> 7, col = e & 127;
                int kg   = k0 + krow;
                int kc   = (kg < Kd) ? kg : (Kd - 1);
                int cg   = cblk + col;  cg = (cg < M) ? cg : (M - 1);
                float v  = X[(size_t)kc * M + cg];
                ((float*)&Xs2[krow >> 1][col])[krow & 1] = (kg < Kd) ? v : 0.f;
            }
        }
        __syncthreads();

        // prefetch all 8 A fragments (b64 each), reused by 4 col tiles
        v2f av[8];
        #pragma unroll
        for (int j = 0; j < 8; ++j)
            av[j] = As2[2 * j + kph][wrow * 16 + mrow];

        const int cb0 = wcol * 64 + mrow;
        #pragma unroll
        for (int j = 0; j < 8; ++j) {
            v2f b0 = Xs2[2 * j + kph][cb0 +  0];
            v2f b1 = Xs2[2 * j + kph][cb0 + 16];
            v2f b2 = Xs2[2 * j + kph][cb0 + 32];
            v2f b3 = Xs2[2 * j + kph][cb0 + 48];
            acc0 = __builtin_amdgcn_wmma_f32_16x16x4_f32(false, av[j], false, b0, (short)0, acc0, false, false);
            acc1 = __builtin_amdgcn_wmma_f32_16x16x4_f32(false, av[j], false, b1, (short)0, acc1, false, false);
            acc2 = __builtin_amdgcn_wmma_f32_16x16x4_f32(false, av[j], false, b2, (short)0, acc2, false, false);
            acc3 = __builtin_amdgcn_wmma_f32_16x16x4_f32(false, av[j], false, b3, (short)0, acc3, false, false);
        }
    }

    const int rbase = rblk + wrow * 16 + ((lane >> 4) << 3); // +0 or +8
    const int cb = cblk + wcol * 64 + mrow;
    v8f accs[4] = {acc0, acc1, acc2, acc3};
    #pragma unroll
    for (int s = 0; s < 4; ++s) {
        int cc = cb + s * 16;
        if (cc < M) {
            #pragma unroll
            for (int v = 0; v < 8; ++v) {
                int rr = rbase + v;
                if (rr < Co) C[(size_t)rr * M + cc] = accs[s][v];
            }
        }
    }
}

// ---------------------------------------------------------------------------
// kNN: per (b,n) keep top-K of pair = -||x_n - x_m||^2  (self included,
// stable tie-break on lower index, matching jax.lax.top_k).
// ---------------------------------------------------------------------------
__global__ __launch_bounds__(256) void knn_kernel(
    const float* __restrict__ pc, int* __restrict__ idx)
{
    int t = blockIdx.x * blockDim.x + threadIdx.x;
    if (t >= Bn * Nn) return;
    int b = t / Nn, n = t % Nn;
    const float* xb = pc + (size_t)b * 3 * Nn;
    float x0 = xb[n], x1 = xb[Nn + n], x2 = xb[2 * Nn + n];
    float vals[Kn];
    int   ids[Kn];
    #pragma unroll
    for (int i = 0; i < Kn; ++i) { vals[i] = -3.4e38f; ids[i] = 0; }
    for (int m = 0; m < Nn; ++m) {
        float d0 = xb[m] - x0, d1 = xb[Nn + m] - x1, d2 = xb[2 * Nn + m] - x2;
        float val = -(d0 * d0 + d1 * d1 + d2 * d2);
        if (val > vals[Kn - 1]) {
            int j = Kn - 1;
            while (j > 0 && vals[j - 1] < val) {
                vals[j] = vals[j - 1]; ids[j] = ids[j - 1]; --j;
            }
            vals[j] = val; ids[j] = m;
        }
    }
    #pragma unroll
    for (int i = 0; i < Kn; ++i) idx[(size_t)t * Kn + i] = ids[i];
}

// ---------------------------------------------------------------------------
// Edge features: feat[c][(b*3+vec)*N*K + n*K + k], channels:
//   c=0: nbr-x, c=1: x, c=2: cross(nbr, x)
// ---------------------------------------------------------------------------
__global__ __launch_bounds__(256) void edge_feat_kernel(
    const float* __restrict__ pc, const int* __restrict__ idx,
    float* __restrict__ feat)
{
    long t = (long)blockIdx.x * blockDim.x + threadIdx.x;
    if (t >= (long)Bn * Nn * Kn) return;
    int k = (int)(t % Kn);
    int n = (int)((t / Kn) % Nn);
    int b = (int)(t / ((long)Nn * Kn));
    const float* xb = pc + (size_t)b * 3 * Nn;
    float x0 = xb[n], x1 = xb[Nn + n], x2 = xb[2 * Nn + n];
    int m = idx[t];
    float p0 = xb[m], p1 = xb[Nn + m], p2 = xb[2 * Nn + m];
    long NK = (long)Nn * Kn;
    long col = (long)b * 3 * NK + (long)n * Kn + k;  // vec=0 column
    feat[0 * MkL + col]          = p0 - x0;
    feat[0 * MkL + col + NK]     = p1 - x1;
    feat[0 * MkL + col + 2 * NK] = p2 - x2;
    feat[1 * MkL + col]          = x0;
    feat[1 * MkL + col + NK]     = x1;
    feat[1 * MkL + col + 2 * NK] = x2;
    feat[2 * MkL + col]          = p1 * x2 - p2 * x1;
    feat[2 * MkL + col + NK]     = p2 * x0 - p0 * x2;
    feat[2 * MkL + col + 2 * NK] = p0 * x1 - p1 * x0;
}

// ---------------------------------------------------------------------------
// VN BatchNorm stats: per channel c, mean/istd of (||vec|| + EPS) over (b, s).
// ---------------------------------------------------------------------------
__global__ __launch_bounds__(256) void vnbn_stats_kernel(
    const float* __restrict__ T, float* __restrict__ mean,
    float* __restrict__ istd, int C, int Bc, int S)
{
    int c = blockIdx.x;
    const float* row = T + (size_t)c * (size_t)Bc * 3 * S;
    int total = Bc * S;
    float s = 0.f, sq = 0.f;
    for (int t = threadIdx.x; t < total; t += blockDim.x) {
        int b = t / S, sp = t % S;
        long base = (long)b * 3 * S + sp;
        float a0 = row[base], a1 = row[base + S], a2 = row[base + 2L * S];
        float nrm = sqrtf(a0 * a0 + a1 * a1 + a2 * a2) + EPSV;
        s += nrm; sq += nrm * nrm;
    }
    __shared__ float sh_s[256], sh_q[256];
    sh_s[threadIdx.x] = s; sh_q[threadIdx.x] = sq;
    __syncthreads();
    for (int o = 128; o > 0; o >>= 1) {
        if ((int)threadIdx.x < o) {
            sh_s[threadIdx.x] += sh_s[threadIdx.x + o];
            sh_q[threadIdx.x] += sh_q[threadIdx.x + o];
        }
        __syncthreads();
    }
    if (threadIdx.x == 0) {
        float mu = sh_s[0] / total;
        float var = sh_q[0] / total - mu * mu;
        mean[c] = mu;
        istd[c] = rsqrtf(var + BNE);
    }
}

// VN LeakyReLU (slope 0) apply
__global__ __launch_bounds__(256) void vn_lrelu_apply_kernel(
    const float* __restrict__ q, const float* __restrict__ d,
    const float* __restrict__ mean, const float* __restrict__ istd,
    const float* __restrict__ g, const float* __restrict__ bt,
    float* __restrict__ out, int C, int Bc, int S)
{
    long t = (long)blockIdx.x * blockDim.x + threadIdx.x;
    long tot = (long)C * Bc * S;
    if (t >= tot) return;
    int c = (int)(t / ((long)Bc * S));
    long rem = t % ((long)Bc * S);
    int b = (int)(rem / S), sp = (int)(rem % S);
    long M = (long)Bc * 3 * S;
    long base = (long)c * M + (long)b * 3 * S + sp;
    float q0 = q[base], q1 = q[base + S], q2 = q[base + 2L * S];
    float d0 = d[base], d1 = d[base + S], d2 = d[base + 2L * S];
    float nrm = sqrtf(q0 * q0 + q1 * q1 + q2 * q2) + EPSV;
    float nb = (nrm - mean[c]) * istd[c] * g[c] + bt[c];
    float sc = nb / nrm;
    q0 *= sc; q1 *= sc; q2 *= sc;
    float dot = q0 * d0 + q1 * d1 + q2 * d2;
    float dsq = d0 * d0 + d1 * d1 + d2 * d2;
    float f = (dot >= 0.f) ? 0.f : dot / (dsq + EPSV);
    out[base]          = q0 - f * d0;
    out[base + S]      = q1 - f * d1;
    out[base + 2L * S] = q2 - f * d2;
}

// VN BatchNorm apply only (conv5 / bn5 path)
__global__ __launch_bounds__(256) void vnbn_apply_kernel(
    const float* __restrict__ q,
    const float* __restrict__ mean, const float* __restrict__ istd,
    const float* __restrict__ g, const float* __restrict__ bt,
    float* __restrict__ out, int C, int Bc, int S)
{
    long t = (long)blockIdx.x * blockDim.x + threadIdx.x;
    long tot = (long)C * Bc * S;
    if (t >= tot) return;
    int c = (int)(t / ((long)Bc * S));
    long rem = t % ((long)Bc * S);
    int b = (int)(rem / S), sp = (int)(rem % S);
    long M = (long)Bc * 3 * S;
    long base = (long)c * M + (long)b * 3 * S + sp;
    float q0 = q[base], q1 = q[base + S], q2 = q[base + 2L * S];
    float nrm = sqrtf(q0 * q0 + q1 * q1 + q2 * q2) + EPSV;
    float nb = (nrm - mean[c]) * istd[c] * g[c] + bt[c];
    float sc = nb / nrm;
    out[base]          = q0 * sc;
    out[base + S]      = q1 * sc;
    out[base + 2L * S] = q2 * sc;
}

// mean over K: [C][B*3*N*K] -> [C][B*3*N]
__global__ __launch_bounds__(256) void mean_over_k_kernel(
    const float* __restrict__ src, float* __restrict__ dst, int C)
{
    long t = (long)blockIdx.x * blockDim.x + threadIdx.x;
    if (t >= (long)C * Mv) return;
    int c = (int)(t / Mv);
    int col = (int)(t % Mv);
    int bv = col / Nn, n = col % Nn;
    const float* p = src + (long)c * MkL + (long)bv * Nn * Kn + (long)n * Kn;
    float s = 0.f;
    #pragma unroll
    for (int k = 0; k < Kn; ++k) s += p[k];
    dst[t] = s * (1.f / Kn);
}

// mean over N: [C][B*3*N] -> [C][B*3]
__global__ __launch_bounds__(256) void mean_over_n_kernel(
    const float* __restrict__ src, float* __restrict__ dst, int C)
{
    int t = blockIdx.x * blockDim.x + threadIdx.x;
    if (t >= C * 12) return;
    int c = t / 12, bv = t % 12;
    const float* p = src + (long)c * Mv + (long)bv * Nn;
    float s = 0.f;
    for (int n = 0; n < Nn; ++n) s += p[n];
    dst[t] = s * (1.f / Nn);
}

// broadcast over N: [C][B*3] -> [C][B*3*N]
__global__ __launch_bounds__(256) void bcast_n_kernel(
    const float* __restrict__ src, float* __restrict__ dst, int C)
{
    long t = (long)blockIdx.x * blockDim.x + threadIdx.x;
    if (t >= (long)C * Mv) return;
    int c = (int)(t / Mv);
    int bv = (int)((t % Mv) / Nn);
    dst[t] = src[c * 12 + bv];
}

// frame transform: dst[(c*3+k)][b*N+n] = sum_j src[c][(b*3+j)*N+n] * z[k][(b*3+j)*N+n]
__global__ __launch_bounds__(256) void transform_kernel(
    const float* __restrict__ src, const float* __restrict__ z,
    float* __restrict__ dst, int Cv)
{
    long t = (long)blockIdx.x * blockDim.x + threadIdx.x;
    if (t >= (long)Cv * Mc) return;
    int c = (int)(t / Mc);
    int rem = (int)(t % Mc);
    int b = rem / Nn, n = rem % Nn;
    long sb = (long)c * Mv + (long)b * 3 * Nn + n;
    float s0 = src[sb], s1 = src[sb + Nn], s2 = src[sb + 2L * Nn];
    #pragma unroll
    for (int k = 0; k < 3; ++k) {
        long zb = (long)k * Mv + (long)b * 3 * Nn + n;
        float v = s0 * z[zb] + s1 * z[zb + Nn] + s2 * z[zb + 2L * Nn];
        dst[(long)(c * 3 + k) * Mc + rem] = v;
    }
}

// row-max over N: src [R][B*N] -> out[r*B + b]
__global__ __launch_bounds__(256) void rowmax_kernel(
    const float* __restrict__ src, float* __restrict__ out)
{
    int r = blockIdx.x, b = blockIdx.y;
    const float* p = src + (long)r * Mc + (long)b * Nn;
    float m = -3.4e38f;
    for (int n = threadIdx.x; n < Nn; n += blockDim.x) m = fmaxf(m, p[n]);
    __shared__ float sh[256];
    sh[threadIdx.x] = m;
    __syncthreads();
    for (int o = 128; o > 0; o >>= 1) {
        if ((int)threadIdx.x < o) sh[threadIdx.x] = fmaxf(sh[threadIdx.x], sh[threadIdx.x + o]);
        __syncthreads();
    }
    if (threadIdx.x == 0) out[r * Bn + b] = sh[0];
}

// expand rows 0..4107 of concat: r<4092 -> rowmax, else label[b][r-4092]
__global__ __launch_bounds__(256) void expand_kernel(
    const float* __restrict__ maxv, const float* __restrict__ label,
    float* __restrict__ concat)
{
    long t = (long)blockIdx.x * blockDim.x + threadIdx.x;
    if (t >= 4108L * Mc) return;
    int r = (int)(t / Mc);
    int rem = (int)(t % Mc);
    int b = rem / Nn;
    concat[t] = (r < 4092) ? maxv[r * Bn + b] : label[b * 16 + (r - 4092)];
}

// plain BN stats over M columns per channel
__global__ __launch_bounds__(256) void bn_stats_kernel(
    const float* __restrict__ h, float* __restrict__ mean,
    float* __restrict__ istd, int C, int M)
{
    int c = blockIdx.x;
    const float* p = h + (long)c * M;
    float s = 0.f, sq = 0.f;
    for (int t = threadIdx.x; t < M; t += blockDim.x) {
        float v = p[t]; s += v; sq += v * v;
    }
    __shared__ float sh_s[256], sh_q[256];
    sh_s[threadIdx.x] = s; sh_q[threadIdx.x] = sq;
    __syncthreads();
    for (int o = 128; o > 0; o >>= 1) {
        if ((int)threadIdx.x < o) {
            sh_s[threadIdx.x] += sh_s[threadIdx.x + o];
            sh_q[threadIdx.x] += sh_q[threadIdx.x + o];
        }
        __syncthreads();
    }
    if (threadIdx.x == 0) {
        float mu = sh_s[0] / M;
        mean[c] = mu;
        istd[c] = rsqrtf(sh_q[0] / M - mu * mu + BNE);
    }
}

// in-place BN + ReLU (conv bias cancels exactly inside batch-norm -> skipped)
__global__ __launch_bounds__(256) void bn_relu_kernel(
    float* __restrict__ h, const float* __restrict__ mean,
    const float* __restrict__ istd, const float* __restrict__ g,
    const float* __restrict__ bt, int C, int M)
{
    long t = (long)blockIdx.x * blockDim.x + threadIdx.x;
    if (t >= (long)C * M) return;
    int c = (int)(t / M);
    float v = (h[t] - mean[c]) * istd[c] * g[c] + bt[c];
    h[t] = fmaxf(v, 0.f);
}

// final: src [50][B*N] + bias -> out [B][50][N]
__global__ __launch_bounds__(256) void out_permute_kernel(
    const float* __restrict__ src, const float* __restrict__ bias,
    float* __restrict__ out)
{
    long t = (long)blockIdx.x * blockDim.x + threadIdx.x;
    if (t >= 50L * Mc) return;
    int o = (int)(t / Mc);
    int rem = (int)(t % Mc);
    int b = rem / Nn, n = rem % Nn;
    out[((long)b * 50 + o) * Nn + n] = src[t] + bias[o];
}

// ---------------------------------------------------------------------------
// Host-side pipeline
// ---------------------------------------------------------------------------
extern "C" void kernel_launch(void* const* d_in, const int* in_sizes, int n_in,
                              void* d_out, int out_size, void* d_ws, size_t ws_size,
                              hipStream_t stream)
{
    (void)in_sizes; (void)n_in; (void)out_size; (void)ws_size;

    const float* pc    = (const float*)d_in[0];
    const float* label = (const float*)d_in[1];
    // 12 VN layers: conv_pos, conv1..3, fstn_conv1..3, fstn_fc1..2, conv4, std_vn1..2
    const float *Wf[12], *Wd[12], *Gg[12], *Bb[12];
    for (int j = 0; j < 12; ++j) {
        Wf[j] = (const float*)d_in[2 + 4 * j + 0];
        Wd[j] = (const float*)d_in[2 + 4 * j + 1];
        Gg[j] = (const float*)d_in[2 + 4 * j + 2];
        Bb[j] = (const float*)d_in[2 + 4 * j + 3];
    }
    const float* fc3W   = (const float*)d_in[50];   // [42,85]
    const float* conv5W = (const float*)d_in[51];   // [682,170]
    const float* bn5g   = (const float*)d_in[52];
    const float* bn5b   = (const float*)d_in[53];
    const float* stdW   = (const float*)d_in[54];   // [3,341]
    const float* c1W = (const float*)d_in[55];
    const float* c1g = (const float*)d_in[57];
    const float* c1t = (const float*)d_in[58];
    const float* c2W = (const float*)d_in[59];
    const float* c2g = (const float*)d_in[61];
    const float* c2t = (const float*)d_in[62];
    const float* c3W = (const float*)d_in[63];
    const float* c3g = (const float*)d_in[65];
    const float* c3t = (const float*)d_in[66];
    const float* c4W = (const float*)d_in[67];
    const float* c4b = (const float*)d_in[68];
    float* outp = (float*)d_out;

    // ---- workspace bump allocator (256B aligned) --------------------------
    size_t off = 0;
    auto alloc = [&](size_t nfloats) -> float* {
        float* p = (float*)((char*)d_ws + off);
        off += nfloats * sizeof(float);
        off = (off + 255) & ~(size_t)255;
        return p;
    };

    int*   idxb    = (int*)alloc(163840);          // [B,N,K]
    float* feat    = alloc(3L * MkL);              // [3][B*3*N*K]
    float* x0      = alloc(21L * Mv);              // conv_pos output after mean_k
    float* buf1234 = alloc(275L * Mv);             // out1|out2|out3|out4 rows
    float* cat42   = alloc(84L * Mv);              // conv4 input
    float* out5cat = alloc(1364L * Mv);            // out5 | mean(out5)
    // --- union block: q/d/z temps, later reused for the 9025-row concat ----
    size_t union_off = off;
    float* qtmp  = alloc(682L * Mv);
    float* dtmp  = alloc(682L * Mv);
    float* zbuf1 = alloc(682L * Mv);
    float* zbuf2 = alloc(341L * Mv);
    float* fA    = alloc(341L * Mv);
    float* fB    = alloc(341L * Mv);
    float* concat = (float*)((char*)d_ws + union_off); // 9025*8192 fits in union
    // -----------------------------------------------------------------------
    float* z3    = alloc(3L * Mv);
    float* tpool = alloc(4096);                    // [341][12]
    float* tf1   = alloc(2048);                    // [170][12]
    float* tf2   = alloc(1024);                    // [85][12]
    float* netg  = alloc(512);                     // [42][12]
    float* statM = alloc(2048);
    float* statI = alloc(2048);
    float* m5    = alloc(8192);                    // [682][12]
    float* maxv  = alloc(4092L * Bn);
    float* h1    = alloc(256L * Mc);
    float* h2    = alloc(256L * Mc);
    float* h3    = alloc(128L * Mc);
    float* h4    = alloc(50L * Mc);

    auto gemm = [&](const float* A, const float* X, float* C, int Co, int Kd, int M) {
        dim3 g((M + 127) / 128, (Co + 63) / 64);
        gemm_wmma_f32<<<g, 256, 0, stream>>>(A, X, C, Co, Kd, M);
    };
    auto vn_lrelu = [&](const float* x, int j, float* out, int Ci, int Co, int S) {
        int M = Bn * 3 * S;
        gemm(Wf[j], x, qtmp, Co, Ci, M);
        gemm(Wd[j], x, dtmp, Co, Ci, M);
        vnbn_stats_kernel<<<Co, 256, 0, stream>>>(qtmp, statM, statI, Co, Bn, S);
        long tot = (long)Co * Bn * S;
        vn_lrelu_apply_kernel<<<(int)((tot + 255) / 256), 256, 0, stream>>>(
            qtmp, dtmp, statM, statI, Gg[j], Bb[j], out, Co, Bn, S);
    };

    // 1) kNN graph + edge features
    knn_kernel<<<(Bn * Nn + 255) / 256, 256, 0, stream>>>(pc, idxb);
    edge_feat_kernel<<<(int)(((long)Bn * Nn * Kn + 255) / 256), 256, 0, stream>>>(pc, idxb, feat);

    // 2) conv_pos over S=N*K, then mean over K
    vn_lrelu(feat, 0, zbuf1, 3, 21, Nn * Kn);
    mean_over_k_kernel<<<(int)((21L * Mv + 255) / 256), 256, 0, stream>>>(zbuf1, x0, 21);

    // 3) conv1..conv3 -> buf1234 rows [0:21), [21:63), [63:105)
    vn_lrelu(x0, 1, buf1234, 21, 21, Nn);
    vn_lrelu(buf1234, 2, buf1234 + 21L * Mv, 21, 42, Nn);
    vn_lrelu(buf1234 + 21L * Mv, 3, buf1234 + 63L * Mv, 42, 42, Nn);

    // 4) fSTN
    vn_lrelu(buf1234 + 63L * Mv, 4, fA, 42, 21, Nn);
    vn_lrelu(fA, 5, fB, 21, 42, Nn);
    vn_lrelu(fB, 6, fA, 42, 341, Nn);
    mean_over_n_kernel<<<(341 * 12 + 255) / 256, 256, 0, stream>>>(fA, tpool, 341);
    vn_lrelu(tpool, 7, tf1, 341, 170, 1);
    vn_lrelu(tf1, 8, tf2, 170, 85, 1);
    gemm(fc3W, tf2, netg, 42, 85, 12);

    // 5) conv4: input = [out3 ; broadcast(netg)]
    hipMemcpyAsync(cat42, buf1234 + 63L * Mv, sizeof(float) * 42L * Mv,
                   hipMemcpyDeviceToDevice, stream);
    bcast_n_kernel<<<(int)((42L * Mv + 255) / 256), 256, 0, stream>>>(netg, cat42 + 42L * Mv, 42);
    vn_lrelu(cat42, 9, buf1234 + 105L * Mv, 84, 170, Nn);

    // 6) conv5 + bn5, then append mean over N -> out5cat [1364][Mv]
    gemm(conv5W, buf1234 + 105L * Mv, qtmp, 682, 170, Mv);
    vnbn_stats_kernel<<<682, 256, 0, stream>>>(qtmp, statM, statI, 682, Bn, Nn);
    vnbn_apply_kernel<<<(int)((682L * Bn * Nn + 255) / 256), 256, 0, stream>>>(
        qtmp, statM, statI, bn5g, bn5b, out5cat, 682, Bn, Nn);
    mean_over_n_kernel<<<(682 * 12 + 255) / 256, 256, 0, stream>>>(out5cat, m5, 682);
    bcast_n_kernel<<<(int)((682L * Mv + 255) / 256), 256, 0, stream>>>(m5, out5cat + 682L * Mv, 682);

    // 7) VNStdFeature: std_vn1, std_vn2, std_lin -> z3 [3][Mv]
    vn_lrelu(out5cat, 10, zbuf1, 1364, 682, Nn);
    vn_lrelu(zbuf1, 11, zbuf2, 682, 341, Nn);
    gemm(stdW, zbuf2, z3, 3, 341, Mv);

    // 8) frame transforms written straight into the concat buffer
    //    rows: [0,4108) expand | [4108,4933) out1234 | [4933,9025) out5f
    transform_kernel<<<(int)((1364L * Mc + 255) / 256), 256, 0, stream>>>(
        out5cat, z3, concat + 4933L * Mc, 1364);
    transform_kernel<<<(int)((275L * Mc + 255) / 256), 256, 0, stream>>>(
        buf1234, z3, concat + 4108L * Mc, 275);
    rowmax_kernel<<<dim3(4092, Bn), 256, 0, stream>>>(concat + 4933L * Mc, maxv);
    expand_kernel<<<(int)((4108L * Mc + 255) / 256), 256, 0, stream>>>(maxv, label, concat);

    // 9) seg head: convs1..convs4
    gemm(c1W, concat, h1, 256, 9025, Mc);
    bn_stats_kernel<<<256, 256, 0, stream>>>(h1, statM, statI, 256, Mc);
    bn_relu_kernel<<<(int)((256L * Mc + 255) / 256), 256, 0, stream>>>(h1, statM, statI, c1g, c1t, 256, Mc);

    gemm(c2W, h1, h2, 256, 256, Mc);
    bn_stats_kernel<<<256, 256, 0, stream>>>(h2, statM, statI, 256, Mc);
    bn_relu_kernel<<<(int)((256L * Mc + 255) / 256), 256, 0, stream>>>(h2, statM, statI, c2g, c2t, 256, Mc);

    gemm(c3W, h2, h3, 128, 256, Mc);
    bn_stats_kernel<<<128, 256, 0, stream>>>(h3, statM, statI, 128, Mc);
    bn_relu_kernel<<<(int)((128L * Mc + 255) / 256), 256, 0, stream>>>(h3, statM, statI, c3g, c3t, 128, Mc);

    gemm(c4W, h3, h4, 50, 128, Mc);
    out_permute_kernel<<<(int)((50L * Mc + 255) / 256), 256, 0, stream>>>(h4, c4b, outp);
}